// Basemodel_15247133901528
// MI455X (gfx1250) — compile-verified
//
#include <hip/hip_runtime.h>
#include <math.h>

// ---------------------------------------------------------------------------
// MPNCOV (iSQRT-COV) for x:[256,256,14,14] fp32 on gfx1250.
// All matmuls run through one WMMA fp32 GEMM core (v_wmma_f32_16x16x4_f32).
// LDS staging is done by the Tensor Data Mover (tensor_load_to_lds), double
// buffered, with TDM zero-fill handling the K=196 tail and TDM LDS padding
// producing the 20-float padded row stride directly.
// Every Newton-Schulz matrix is symmetric (polynomial in A), so the B operand
// panel is always Bs[n][k] = Bmem[n*ldb+k] (plain 2D tile copy, no transpose)
// and WMMA computes A @ B exactly.
// ---------------------------------------------------------------------------

constexpr int C    = 256;          // channels
constexpr int MSP  = 196;          // spatial positions (14*14)
constexpr int NB   = 256;          // batch
constexpr int MAT  = C * C;        // 65536 elements per matrix
constexpr int TRI  = C * (C + 1) / 2;  // 32896 triu elements

constexpr int BM = 64;             // block tile M
constexpr int BN = 128;            // block tile N
constexpr int KC = 16;             // K panel depth (dwords per row)
constexpr int LDSP = 20;           // padded LDS row stride (floats) = 16 + 4 pad

typedef float v2f __attribute__((ext_vector_type(2)));
typedef float v8f __attribute__((ext_vector_type(8)));
typedef unsigned int u32x4 __attribute__((ext_vector_type(4)));
typedef unsigned int u32x8 __attribute__((ext_vector_type(8)));

// ---------------------------------------------------------------------------
// Issue one TDM 2D tile load: rows x KC fp32 tile, row stride ldElems, into
// LDS at lds_byte_off with 4-dword padding after every 16 dwords (-> LDSP=20).
// remCols (= tensor_dim0) < KC zero-fills the K tail via TDM OOB handling.
// D# group0/group1 per CDNA5 ISA 8.3/8.4; 2D tensor -> 2-group form.
// ---------------------------------------------------------------------------
static __device__ __forceinline__ void tdm_issue(const float* gsrc,
                                                 unsigned int lds_byte_off,
                                                 unsigned int remCols,
                                                 unsigned int rows,
                                                 unsigned int ldElems) {
    const unsigned long long ga = (unsigned long long)(uintptr_t)gsrc;
    u32x4 g0;
    g0[0] = 1u;                                       // count=1, user desc
    g0[1] = lds_byte_off;                             // lds_addr (bytes)
    g0[2] = (unsigned int)ga;                         // global_addr[31:0]
    g0[3] = ((unsigned int)(ga >> 32) & 0x01FFFFFFu)  // global_addr[56:32]
            | (2u << 30);                             // type=2 (image)
    u32x8 g1;
    // data_size=2 (4B) | pad_enable | pad_interval=3 (16 dw) | pad_amount=3 (4 dw)
    g1[0] = (2u << 16) | (1u << 20) | (3u << 22) | (3u << 25);
    g1[1] = (remCols & 0xFFFFu) << 16;                // tensor_dim0[15:0]
    g1[2] = (remCols >> 16) | ((rows & 0xFFFFu) << 16); // tensor_dim0[31:16] | tensor_dim1[15:0]
    g1[3] = (rows >> 16) | ((unsigned int)KC << 16);  // tensor_dim1[31:16] | tile_dim0=16
    g1[4] = rows;                                     // tile_dim1 (tile_dim2=0)
    g1[5] = ldElems;                                  // tensor_dim0_stride[31:0]
    g1[6] = 0u;                                       // stride[47:32] | tensor_dim1_stride lo
    g1[7] = 0u;
    asm volatile("tensor_load_to_lds %0, %1" :: "s"(g0), "s"(g1) : "memory");
}

// ---------------------------------------------------------------------------
// Per-channel spatial means: mu[b,c] = mean_m x[b,c,m]
// ---------------------------------------------------------------------------
__global__ __launch_bounds__(256) void means_kernel(const float* __restrict__ x,
                                                    float* __restrict__ mu) {
    const int b = blockIdx.x;
    const int c = threadIdx.x;
    const float4* xp = reinterpret_cast<const float4*>(x + ((size_t)b * C + c) * MSP);
    float s = 0.f;
    #pragma unroll 7
    for (int q = 0; q < MSP / 4; ++q) {  // 49 float4 loads
        float4 v = xp[q];
        s += v.x + v.y + v.z + v.w;
    }
    mu[b * C + c] = s * (1.0f / MSP);
}

// ---------------------------------------------------------------------------
// trace(cov) per batch -> invN = 1/tr, sqrtN = sqrt(tr)
// ---------------------------------------------------------------------------
__global__ __launch_bounds__(256) void trace_kernel(const float* __restrict__ cov,
                                                    float* __restrict__ invN,
                                                    float* __restrict__ sqrtN) {
    __shared__ float red[256];
    const int b = blockIdx.x;
    const int t = threadIdx.x;
    red[t] = cov[(size_t)b * MAT + (size_t)t * (C + 1)];
    __syncthreads();
    #pragma unroll
    for (int s = 128; s > 0; s >>= 1) {
        if (t < s) red[t] += red[t + s];
        __syncthreads();
    }
    if (t == 0) {
        float tr = red[0];
        invN[b]  = 1.0f / tr;
        sqrtN[b] = sqrtf(tr);
    }
}

// ---------------------------------------------------------------------------
// WMMA fp32 GEMM core: D = epilogue( A @ Bmem^T )  (Bmem symmetric => A@B)
//   A   : row-major [C x K], stride lda, batch stride aStride
//   Bmem: row-major, element B[k][n] = Bmem[n*ldb + k]
// MODE 0: cov     = acc/196 - mu_i*mu_j
// MODE 1: acc = cov@cov; s=invN:  D = 1.5 s cov - 0.5 s^2 acc ; D2 = 1.5I - 0.5 s cov
// MODE 2: D = 1.5*I - 0.5*acc
// MODE 3: D = acc
// MODE 4: out_triu = acc * sqrtN  (packed upper-triangle scatter)
// ---------------------------------------------------------------------------
template <int MODE>
__global__ __launch_bounds__(256) void gemm_ns(
    const float* __restrict__ A, int lda, int aStride,
    const float* __restrict__ Bmem, int ldb, int bStride,
    float* __restrict__ D, float* __restrict__ D2,
    int K,
    const float* __restrict__ covbuf,
    const float* __restrict__ invN,
    const float* __restrict__ sqrtN,
    const float* __restrict__ mu,
    float* __restrict__ out)
{
    __shared__ float As[2][BM * LDSP];   // double-buffered A panel [BM][KC] padded
    __shared__ float Bs[2][BN * LDSP];   // double-buffered B panel [BN][KC] padded

    const int tid  = threadIdx.x;
    const int lane = tid & 31;
    const int wave = tid >> 5;
    const int l16  = lane & 15;
    const int lh   = lane >> 4;        // lane half: 0 or 1
    const int wm   = (wave >> 2) * 32; // wave row offset in block tile (0 / 32)
    const int wn   = (wave & 3) * 32;  // wave col offset (0/32/64/96)

    const int b      = blockIdx.z;
    const int blockM = blockIdx.y * BM;
    const int blockN = blockIdx.x * BN;

    const float* Apanel = A    + (size_t)b * aStride + (size_t)blockM * lda;
    const float* Bpanel = Bmem + (size_t)b * bStride + (size_t)blockN * ldb;

    const unsigned int asOff[2] = { (unsigned int)(uintptr_t)&As[0][0],
                                    (unsigned int)(uintptr_t)&As[1][0] };
    const unsigned int bsOff[2] = { (unsigned int)(uintptr_t)&Bs[0][0],
                                    (unsigned int)(uintptr_t)&Bs[1][0] };

    const v8f zero = {0, 0, 0, 0, 0, 0, 0, 0};
    v8f acc[2][2];
    #pragma unroll
    for (int im = 0; im < 2; ++im)
        #pragma unroll
        for (int in = 0; in < 2; ++in) acc[im][in] = zero;

    const int nChunks = (K + KC - 1) / KC;

    // Prologue: wave0 DMAs chunk 0 into buffer 0.
    if (wave == 0) {
        const unsigned int rem0 = (K < KC) ? (unsigned int)K : (unsigned int)KC;
        tdm_issue(Apanel, asOff[0], rem0, BM, (unsigned int)lda);
        tdm_issue(Bpanel, bsOff[0], rem0, BN, (unsigned int)ldb);
    }

    for (int ic = 0; ic < nChunks; ++ic) {
        if (wave == 0) __builtin_amdgcn_s_wait_tensorcnt(0);  // chunk ic landed
        __syncthreads();                                      // publish buf[ic&1]
        if (wave == 0 && (ic + 1) < nChunks) {                // prefetch chunk ic+1
            const int kc = (ic + 1) * KC;
            int rem = K - kc; if (rem > KC) rem = KC;
            tdm_issue(Apanel + kc, asOff[(ic + 1) & 1], (unsigned int)rem, BM, (unsigned int)lda);
            tdm_issue(Bpanel + kc, bsOff[(ic + 1) & 1], (unsigned int)rem, BN, (unsigned int)ldb);
        }

        const float* as = &As[ic & 1][0];
        const float* bs = &Bs[ic & 1][0];

        // ---- 4 k-steps of V_WMMA_F32_16X16X4_F32 per panel ----
        #pragma unroll
        for (int ks = 0; ks < KC; ks += 4) {
            v2f afrag[2], bfrag[2];
            // A 16x4 fragment: lane l16 = M row; VGPR0/1 = K {ks+2*lh, ks+2*lh+1}
            #pragma unroll
            for (int im = 0; im < 2; ++im)
                afrag[im] = *reinterpret_cast<const v2f*>(
                    &as[(wm + im * 16 + l16) * LDSP + ks + 2 * lh]);
            // B 4x16 fragment: lane l16 = N col; same K split as A
            #pragma unroll
            for (int in = 0; in < 2; ++in)
                bfrag[in] = *reinterpret_cast<const v2f*>(
                    &bs[(wn + in * 16 + l16) * LDSP + ks + 2 * lh]);
            #pragma unroll
            for (int im = 0; im < 2; ++im)
                #pragma unroll
                for (int in = 0; in < 2; ++in)
                    acc[im][in] = __builtin_amdgcn_wmma_f32_16x16x4_f32(
                        false, afrag[im], false, bfrag[in],
                        (short)0, acc[im][in], false, false);
        }
    }

    // ---- fused epilogue ----
    float s1 = 0.f, sq = 0.f;
    if (MODE == 1) s1 = invN[b];
    if (MODE == 4) sq = sqrtN[b];

    #pragma unroll
    for (int im = 0; im < 2; ++im) {
        #pragma unroll
        for (int in = 0; in < 2; ++in) {
            #pragma unroll
            for (int g = 0; g < 8; ++g) {
                const int i = blockM + wm + im * 16 + g + 8 * lh;  // C/D: M = g (+8 for hi half)
                const int j = blockN + wn + in * 16 + l16;         // N = lane%16
                const float v = acc[im][in][g];
                if constexpr (MODE == 0) {
                    const float mi = mu[b * C + i];
                    const float mj = mu[b * C + j];
                    D[(size_t)b * MAT + i * C + j] = v * (1.0f / MSP) - mi * mj;
                } else if constexpr (MODE == 1) {
                    const float cv = covbuf[(size_t)b * MAT + i * C + j];
                    D [(size_t)b * MAT + i * C + j] = 1.5f * s1 * cv - 0.5f * s1 * s1 * v;
                    D2[(size_t)b * MAT + i * C + j] = ((i == j) ? 1.5f : 0.0f) - 0.5f * s1 * cv;
                } else if constexpr (MODE == 2) {
                    D[(size_t)b * MAT + i * C + j] = ((i == j) ? 1.5f : 0.0f) - 0.5f * v;
                } else if constexpr (MODE == 3) {
                    D[(size_t)b * MAT + i * C + j] = v;
                } else {  // MODE == 4: scaled triu scatter
                    if (i <= j) {
                        const int idx = i * C - (i * (i - 1)) / 2 + (j - i);
                        out[(size_t)b * TRI + idx] = v * sq;
                    }
                }
            }
        }
    }
}

// ---------------------------------------------------------------------------
// Launch schedule (stream-ordered; intermediates stay hot in 192MB L2):
//   means -> cov(GEMM,K=196) -> trace
//   G1: P=cov@cov -> Y1(mat1), Z1(mat2)
//   G2: Z1@Y1 -> ZY1(mat0)      G3: Y1@ZY1 -> Y2(mat3)
//   G4: ZY1@Z1 -> Z2(mat1)      G5: Z2@Y2 -> U(mat2)
//   G6: Y2@U * sqrt(tr) -> triu(out)
// ---------------------------------------------------------------------------
extern "C" void kernel_launch(void* const* d_in, const int* in_sizes, int n_in,
                              void* d_out, int out_size, void* d_ws, size_t ws_size,
                              hipStream_t stream) {
    const float* x = (const float*)d_in[0];
    float* out = (float*)d_out;

    float* wsf   = (float*)d_ws;
    float* mat0  = wsf;                               // cov -> ZY1
    float* mat1  = mat0 + (size_t)NB * MAT;           // Y1 -> Z2
    float* mat2  = mat1 + (size_t)NB * MAT;           // Z1 -> U
    float* mat3  = mat2 + (size_t)NB * MAT;           // Y2
    float* mu    = mat3 + (size_t)NB * MAT;           // [NB*C]
    float* invN  = mu + (size_t)NB * C;               // [NB]
    float* sqrtN = invN + NB;                         // [NB]

    const dim3 blk(256);
    const dim3 grd(C / BN, C / BM, NB);               // (2, 4, 256)

    means_kernel<<<NB, 256, 0, stream>>>(x, mu);

    // cov = Gram(x)/196 - mu mu^T   (K = 196, both operands are x)
    gemm_ns<0><<<grd, blk, 0, stream>>>(x, MSP, C * MSP, x, MSP, C * MSP,
                                        mat0, nullptr, MSP,
                                        nullptr, nullptr, nullptr, mu, nullptr);

    trace_kernel<<<NB, 256, 0, stream>>>(mat0, invN, sqrtN);

    // G1: Y1 = 1.5 s cov - 0.5 s^2 cov^2 ; Z1 = 1.5I - 0.5 s cov
    gemm_ns<1><<<grd, blk, 0, stream>>>(mat0, C, MAT, mat0, C, MAT,
                                        mat1, mat2, C,
                                        mat0, invN, nullptr, nullptr, nullptr);
    // G2: ZY1 = 1.5I - 0.5 (Z1 @ Y1)
    gemm_ns<2><<<grd, blk, 0, stream>>>(mat2, C, MAT, mat1, C, MAT,
                                        mat0, nullptr, C,
                                        nullptr, nullptr, nullptr, nullptr, nullptr);
    // G3: Y2 = Y1 @ ZY1
    gemm_ns<3><<<grd, blk, 0, stream>>>(mat1, C, MAT, mat0, C, MAT,
                                        mat3, nullptr, C,
                                        nullptr, nullptr, nullptr, nullptr, nullptr);
    // G4: Z2 = ZY1 @ Z1
    gemm_ns<3><<<grd, blk, 0, stream>>>(mat0, C, MAT, mat2, C, MAT,
                                        mat1, nullptr, C,
                                        nullptr, nullptr, nullptr, nullptr, nullptr);
    // G5: U = 1.5I - 0.5 (Z2 @ Y2)
    gemm_ns<2><<<grd, blk, 0, stream>>>(mat1, C, MAT, mat3, C, MAT,
                                        mat2, nullptr, C,
                                        nullptr, nullptr, nullptr, nullptr, nullptr);
    // G6: out_triu = (Y2 @ U) * sqrt(trace)
    gemm_ns<4><<<grd, blk, 0, stream>>>(mat3, C, MAT, mat2, C, MAT,
                                        nullptr, nullptr, C,
                                        nullptr, nullptr, sqrtN, nullptr, out);
}